// PositionMagnitude_4741643894786
// MI455X (gfx1250) — compile-verified
//
#include <hip/hip_runtime.h>

// Problem constants (match reference)
namespace {
constexpr int kNL = 90, kNB = 20, kNMU = 46, kNLF = 51;
constexpr int kCells = kNL * kNB * kNMU;   // 82800
constexpr int kRows  = kNL * kNB;          // 1800  (l*b rows)
constexpr int kNOut  = kNLF - kNMU + 1;    // 6
constexpr int kTiles = (kRows + 15) / 16;  // 113
}

typedef __attribute__((ext_vector_type(2))) float v2f;
typedef __attribute__((ext_vector_type(8))) float v8f;

// ---------------------------------------------------------------------------
// Stage 0: zero the histogram scratch (d_ws) every call (harness poisons ws).
// ---------------------------------------------------------------------------
__global__ void zero_hist_kernel(float* __restrict__ hist) {
  int i = blockIdx.x * blockDim.x + threadIdx.x;
  if (i < kCells) hist[i] = 0.0f;
}

// ---------------------------------------------------------------------------
// Stage 1: mass-weighted 3D histogram, direct L2-resident float atomics.
// idx = rint((x - min)/dx)  with dx as exact f32 division constants (matches
// XLA's f32 divide + round-half-even). All-valid check, then one
// device-scope global_atomic_add_f32 (no return -> STOREcnt, fire-and-forget
// into the device-coherent L2; SCOPE_DEV guarantees cross-WGP RMW coherence).
// ---------------------------------------------------------------------------
__global__ __launch_bounds__(256) void hist_kernel(
    const float* __restrict__ lbm, const float* __restrict__ mass,
    float* __restrict__ hist, int n) {
  const float mn0 = -90.0f, mn1 = -12.0f, mn2 = 7.0f;
  const float dx0 = 180.0f / 89.0f;  // (90-(-90))/(90-1)
  const float dx1 = 24.0f / 19.0f;   // (12-(-12))/(20-1)
  const float dx2 = 9.0f / 45.0f;    // (16-7)/(46-1) = 0.2f
  const int stride = (int)(gridDim.x * blockDim.x);
  for (int i = blockIdx.x * blockDim.x + threadIdx.x; i < n; i += stride) {
    float x0 = lbm[3 * i + 0];
    float x1 = lbm[3 * i + 1];
    float x2 = lbm[3 * i + 2];
    int i0 = (int)__builtin_rintf((x0 - mn0) / dx0);  // v_rndne: half-to-even
    int i1 = (int)__builtin_rintf((x1 - mn1) / dx1);
    int i2 = (int)__builtin_rintf((x2 - mn2) / dx2);
    if ((unsigned)i0 < (unsigned)kNL && (unsigned)i1 < (unsigned)kNB &&
        (unsigned)i2 < (unsigned)kNMU) {
      float w = mass[i];
      float* p = hist + ((i0 * kNB + i1) * kNMU + i2);
      // Native HW float atomic, device scope (agent-coherent RMW in L2).
      // EXEC-predicated by the surrounding divergent branch.
      asm volatile("global_atomic_add_f32 %0, %1, off scope:SCOPE_DEV"
                   :: "v"(p), "v"(w)
                   : "memory");
    }
  }
}

// ---------------------------------------------------------------------------
// Stage 2: out[row, k] = sum_m hist[row, m] * lf[k + 45 - m]
// via V_WMMA_F32_16X16X4_F32. One wave (32 lanes) per 16-row tile.
// K = 46 padded to 48 -> 12 chained WMMAs accumulating in v8f.
//
// ISA VGPR layouts (CDNA5 ISA):
//   A 16x4 f32 : lanes 0-15 -> {K=0 (a.x), K=1 (a.y)}, lanes 16-31 -> {K=2,K=3}
//   B 4x16 f32 : lanes 0-15 -> N=0..15 of rows {K=0 (b.x), K=1 (b.y)},
//                lanes 16-31 -> rows {K=2, K=3}
//   C/D 16x16  : VGPR r: lanes 0-15 -> (M=r, N=lane), lanes 16-31 -> (M=r+8)
// ---------------------------------------------------------------------------
__global__ __launch_bounds__(32) void conv_wmma_kernel(
    const float* __restrict__ hist, const float* __restrict__ lf,
    float* __restrict__ out) {
  const int tile = blockIdx.x;          // 0..112
  const int lane = threadIdx.x;         // 0..31
  const int half = lane >> 4;           // 0 or 1
  const int l16  = lane & 15;
  const int row  = tile * 16 + l16;     // A-matrix row for this lane

  v8f c = {};
  #pragma unroll
  for (int kc = 0; kc < 12; ++kc) {
    const int ka = kc * 4 + half * 2;   // K index of {a.x,b.x}; +1 -> {a.y,b.y}
    v2f a, b;
    // A = hist tile (zero-padded for row >= 1800 or K >= 46)
    a.x = (row < kRows && ka     < kNMU) ? hist[row * kNMU + ka]     : 0.0f;
    a.y = (row < kRows && ka + 1 < kNMU) ? hist[row * kNMU + ka + 1] : 0.0f;
    // B[k][n] = lf[n + 45 - k] for n < 6, k < 46, else 0
    const int n = l16;
    b.x = (n < kNOut && ka     < kNMU) ? lf[n + (kNMU - 1) - ka]       : 0.0f;
    b.y = (n < kNOut && ka + 1 < kNMU) ? lf[n + (kNMU - 1) - (ka + 1)] : 0.0f;
    c = __builtin_amdgcn_wmma_f32_16x16x4_f32(
        /*neg_a=*/false, a, /*neg_b=*/false, b,
        /*c_mod=*/(short)0, c, /*reuse_a=*/false, /*reuse_b=*/false);
  }

  #pragma unroll
  for (int r = 0; r < 8; ++r) {
    const int m = tile * 16 + r + half * 8;
    const int n = l16;
    if (m < kRows && n < kNOut) out[m * kNOut + n] = c[r];
  }
}

// ---------------------------------------------------------------------------
extern "C" void kernel_launch(void* const* d_in, const int* in_sizes, int n_in,
                              void* d_out, int out_size, void* d_ws, size_t ws_size,
                              hipStream_t stream) {
  const float* lbm  = (const float*)d_in[0];  // [N,3] f32
  const float* mass = (const float*)d_in[1];  // [N]   f32
  const float* lf   = (const float*)d_in[2];  // [51]  f32
  float* out  = (float*)d_out;                // [1800*6] f32
  float* hist = (float*)d_ws;                 // [82800] f32 scratch

  const int n = in_sizes[1];                  // particle count

  zero_hist_kernel<<<(kCells + 255) / 256, 256, 0, stream>>>(hist);

  int blocks = (n + 255) / 256;
  if (blocks > 16384) blocks = 16384;         // grid-stride persistent-ish
  hist_kernel<<<blocks, 256, 0, stream>>>(lbm, mass, hist, n);

  conv_wmma_kernel<<<kTiles, 32, 0, stream>>>(hist, lf, out);
}